// SamplingLayer_27608049778808
// MI455X (gfx1250) — compile-verified
//
#include <hip/hip_runtime.h>

// ---------------------------------------------------------------------------
// Farthest Point Sampling for MI455X (gfx1250, wave32, WGP, 320KB LDS, TDM)
//   points: (16, 65536, 3) f32  ->  indices: (16, 2048) i32
// One 1024-thread block (32 waves = 1 full WGP) per batch.
// min_dist[65536] lives in LDS (256KB) -- only possible with CDNA5's 320KB.
// Final index array is DMA'd LDS->global with the Tensor Data Mover, then
// redundantly stored by wave 0 as a safety net (idempotent if TDM is exact).
// ---------------------------------------------------------------------------

#define NPTS   65536
#define NSAMP  2048
#define NTHR   1024
#define NGRP   (NPTS / 4)        // 4-point groups (48B, 16B-aligned)
#define GPT    (NGRP / NTHR)     // 16 groups per thread
#define BIGF   1e10f

typedef unsigned int v4u __attribute__((ext_vector_type(4)));
typedef int          v8i __attribute__((ext_vector_type(8)));
typedef int          v4i __attribute__((ext_vector_type(4)));

struct SMem {
    float md[NPTS];        // 256KB  min squared distance per point
    int   idx[NSAMP];      // 8KB    selected indices (TDM'd out at the end)
    float redv[32];        // per-wave argmax values
    int   redi[32];        // per-wave argmax indices
    float c[4];            // broadcast centroid (x,y,z)
};

__device__ __forceinline__ float sqd(float dx, float dy, float dz) {
    return fmaf(dz, dz, fmaf(dy, dy, dx * dx));
}

__global__ __launch_bounds__(NTHR) void fps_kernel(const float* __restrict__ pts,
                                                   int* __restrict__ out)
{
    __shared__ SMem sm;

    const int b   = blockIdx.x;
    const int tid = threadIdx.x;
    const float*  P   = pts + (size_t)b * NPTS * 3;
    const float4* Pf4 = reinterpret_cast<const float4*>(P);   // 16B-aligned

    // Init min-dist to BIG (reference seeds with 1e10), seed index 0.
    for (int i = tid; i < NPTS; i += NTHR) sm.md[i] = BIGF;
    if (tid == 0) {
        sm.idx[0] = 0;
        sm.c[0] = P[0]; sm.c[1] = P[1]; sm.c[2] = P[2];
    }
    __syncthreads();

    float cx = sm.c[0], cy = sm.c[1], cz = sm.c[2];

    for (int it = 1; it < NSAMP; ++it) {
        float bestv = -1.0f;
        int   besti = 0;

        // 65536 points: 16 groups of 4 per thread. Coalesced b128 global
        // loads + b128 LDS read-modify-write of min_dist. Ascending index
        // order + strict '>' gives first-max (jnp.argmax) tiebreak.
        #pragma unroll 8
        for (int k = 0; k < GPT; ++k) {
            const int g  = tid + k * NTHR;
            const int p0 = g << 2;
            const float4 v0 = Pf4[3 * g + 0];
            const float4 v1 = Pf4[3 * g + 1];
            const float4 v2 = Pf4[3 * g + 2];

            float4 m = *reinterpret_cast<float4*>(&sm.md[p0]);

            const float d0 = sqd(v0.x - cx, v0.y - cy, v0.z - cz);
            const float d1 = sqd(v0.w - cx, v1.x - cy, v1.y - cz);
            const float d2 = sqd(v1.z - cx, v1.w - cy, v2.x - cz);
            const float d3 = sqd(v2.y - cx, v2.z - cy, v2.w - cz);

            m.x = fminf(m.x, d0);
            m.y = fminf(m.y, d1);
            m.z = fminf(m.z, d2);
            m.w = fminf(m.w, d3);

            *reinterpret_cast<float4*>(&sm.md[p0]) = m;

            if (m.x > bestv) { bestv = m.x; besti = p0;     }
            if (m.y > bestv) { bestv = m.y; besti = p0 + 1; }
            if (m.z > bestv) { bestv = m.z; besti = p0 + 2; }
            if (m.w > bestv) { bestv = m.w; besti = p0 + 3; }
        }

        // Wave32 shuffle reduction (max value, min index on ties).
        #pragma unroll
        for (int off = 16; off > 0; off >>= 1) {
            float ov = __shfl_down(bestv, off, 32);
            int   oi = __shfl_down(besti, off, 32);
            if (ov > bestv || (ov == bestv && oi < besti)) { bestv = ov; besti = oi; }
        }
        if ((tid & 31) == 0) { sm.redv[tid >> 5] = bestv; sm.redi[tid >> 5] = besti; }
        __syncthreads();

        // Cross-wave reduction in wave 0 (32 waves -> 32 lanes).
        if (tid < 32) {
            bestv = sm.redv[tid];
            besti = sm.redi[tid];
            #pragma unroll
            for (int off = 16; off > 0; off >>= 1) {
                float ov = __shfl_down(bestv, off, 32);
                int   oi = __shfl_down(besti, off, 32);
                if (ov > bestv || (ov == bestv && oi < besti)) { bestv = ov; besti = oi; }
            }
            if (tid == 0) {
                sm.idx[it] = besti;
                const float* cp = P + (size_t)besti * 3;   // L2-hot
                sm.c[0] = cp[0]; sm.c[1] = cp[1]; sm.c[2] = cp[2];
            }
        }
        __syncthreads();
        cx = sm.c[0]; cy = sm.c[1]; cz = sm.c[2];
    }

    __syncthreads();

    // ---- Epilogue: DMA the 2048 int32 indices LDS -> global with the TDM ---
    if ((tid >> 5) == 0) {   // wave 0 only (uniform branch; TDM ignores EXEC)
#if __has_builtin(__builtin_amdgcn_tensor_store_from_lds) && __has_builtin(__builtin_amdgcn_s_wait_tensorcnt)
        const unsigned long long ga =
            (unsigned long long)(out + (size_t)b * NSAMP);
        // Flat LDS pointer: addr[31:0] is the wave-relative LDS byte offset.
        const unsigned lds_off =
            (unsigned)(unsigned long long)(const void*)&sm.idx[0];

        // D# group 0: count=1 | lds_addr | global_addr[56:0] | type=2
        v4u g0;
        g0.x = 1u;
        g0.y = lds_off;
        g0.z = (unsigned)(ga & 0xFFFFFFFFu);
        g0.w = ((unsigned)((ga >> 32) & 0x01FFFFFFu)) | (2u << 30);

        // D# group 1: data_size=4B(2); tensor_dim0=2048; tensor_dim1=1;
        //             tile_dim0=2048; tile_dim1=1; tensor_dim0_stride=2048.
        // OOB rule clamps all writes to tensor_dim0 x tensor_dim1 elements,
        // so the TDM cannot touch memory outside this batch's 8KB output.
        v8i g1;
        g1[0] = 0x00020000;            // data_size=2 (4 bytes), wg_mask=0
        g1[1] = (int)(2048u << 16);    // tensor_dim0[15:0] @ bits 63:48
        g1[2] = 0x00010000;            // tensor_dim0[31:16]=0, tensor_dim1=1
        g1[3] = (int)(2048u << 16);    // tile_dim0 @ bits 127:112
        g1[4] = 0x00000001;            // tile_dim1=1, tile_dim2=0
        g1[5] = 2048;                  // tensor_dim0_stride[31:0]
        g1[6] = 0;                     // stride hi / tensor_dim1_stride lo
        g1[7] = 0;

        v4i g2 = {0, 0, 0, 0};         // 2D tensor: groups 2/3 unused
        v4i g3 = {0, 0, 0, 0};
        v8i g4 = {0, 0, 0, 0, 0, 0, 0, 0};  // extra group (clang-23 6-arg form)

        __builtin_amdgcn_tensor_store_from_lds(g0, g1, g2, g3, g4, 0);
        __builtin_amdgcn_s_wait_tensorcnt(0);
#endif
        // Safety net: redundant plain store of the same values AFTER the TDM
        // completes. Idempotent when the TDM descriptor is exact; guarantees
        // a correct d_out even if the TDM op were to NOP.
        for (int i = tid; i < NSAMP; i += 32)
            out[(size_t)b * NSAMP + i] = sm.idx[i];
    }
}

extern "C" void kernel_launch(void* const* d_in, const int* in_sizes, int n_in,
                              void* d_out, int out_size, void* d_ws, size_t ws_size,
                              hipStream_t stream) {
    (void)n_in; (void)out_size; (void)d_ws; (void)ws_size;
    const float* pts = (const float*)d_in[0];
    int*         out = (int*)d_out;
    const int nb = in_sizes[0] / (NPTS * 3);   // = 16 batches
    fps_kernel<<<dim3(nb), dim3(NTHR), 0, stream>>>(pts, out);
}